// EffdetNMS_77747497992599
// MI455X (gfx1250) — compile-verified
//
#include <hip/hip_runtime.h>
#include <hip/hip_bf16.h>
#include <cfloat>

// Problem constants (from reference setup_inputs)
#define NB       8
#define NDET     4096
#define MAXDET   100
#define NTHREADS 512
#define KPT      (NDET / NTHREADS)   // 8 elements per thread
#define NWAVES   (NTHREADS / 32)     // 16 wave32s
#define IOU_THR  0.5f

typedef int i32x4 __attribute__((ext_vector_type(4)));
typedef int i32x8 __attribute__((ext_vector_type(8)));

// Candidate selection tuple carried through the argmax reduction.
struct Sel { float s; int i; float x1, y1, x2, y2; int c; };

__device__ __forceinline__ void waveReduceSel(Sel& v) {
#pragma unroll
    for (int d = 16; d > 0; d >>= 1) {
        Sel o;
        o.s  = __shfl_down(v.s,  d, 32);
        o.i  = __shfl_down(v.i,  d, 32);
        o.x1 = __shfl_down(v.x1, d, 32);
        o.y1 = __shfl_down(v.y1, d, 32);
        o.x2 = __shfl_down(v.x2, d, 32);
        o.y2 = __shfl_down(v.y2, d, 32);
        o.c  = __shfl_down(v.c,  d, 32);
        // max score; tie-break lower original index (matches stable argsort(-scores))
        bool take = (o.s > v.s) || (o.s == v.s && o.i >= 0 && (v.i < 0 || o.i < v.i));
        if (take) v = o;
    }
}

// LDS scratch layout (floats): [0..15] wave scores, [16..31] wave idx (int),
// [32..47] x1, [48..63] y1, [64..79] x2, [80..95] y2, [96..111] cls (int),
// [112..118] broadcast {s, i, x1, y1, x2, y2, c}
__device__ __forceinline__ void blockReduceBroadcast(Sel v, int lane, int wave, float* sm) {
    int* smi = reinterpret_cast<int*>(sm);
    waveReduceSel(v);
    if (lane == 0) {
        sm[wave]       = v.s;
        smi[16 + wave] = v.i;
        sm[32 + wave]  = v.x1;
        sm[48 + wave]  = v.y1;
        sm[64 + wave]  = v.x2;
        sm[80 + wave]  = v.y2;
        smi[96 + wave] = v.c;
    }
    __syncthreads();
    if (wave == 0) {
        Sel r;
        if (lane < NWAVES) {
            r.s  = sm[lane];
            r.i  = smi[16 + lane];
            r.x1 = sm[32 + lane];
            r.y1 = sm[48 + lane];
            r.x2 = sm[64 + lane];
            r.y2 = sm[80 + lane];
            r.c  = smi[96 + lane];
        } else {
            r.s = -FLT_MAX; r.i = -1; r.x1 = r.y1 = r.x2 = r.y2 = 0.0f; r.c = 0;
        }
        waveReduceSel(r);
        if (lane == 0) {
            sm[112]  = r.s;
            smi[113] = r.i;
            sm[114]  = r.x1;
            sm[115]  = r.y1;
            sm[116]  = r.x2;
            sm[117]  = r.y2;
            smi[118] = r.c;
        }
    }
    __syncthreads();
}

__global__ __launch_bounds__(NTHREADS)
void effdet_nms_kernel(const float* __restrict__ cls_o,
                       const float* __restrict__ box_o,
                       const int*   __restrict__ indices,
                       const int*   __restrict__ classes,
                       const float* __restrict__ anchors,
                       const float* __restrict__ img_scale,
                       const float* __restrict__ img_size,
                       float*       __restrict__ out)
{
    // 64 KB staging buffer for raw box deltas (TDM target); reused as reduce
    // scratch after the decode phase.
    __shared__ __align__(16) unsigned char smem[NDET * 4 * sizeof(float)];
    float* rawBox = reinterpret_cast<float*>(smem);

    const int b    = blockIdx.x;
    const int tid  = threadIdx.x;
    const int lane = tid & 31;
    const int wave = tid >> 5;

    // ---- Phase 0: Tensor Data Mover — DMA box_outputs[b] (4096x4 f32 = 64KB)
    // straight into LDS, issued by wave 0 only. ----
    if (wave == 0) {
        unsigned lds_addr = (unsigned)(unsigned long long)(size_t)(void*)rawBox;
        unsigned long long ga =
            (unsigned long long)(size_t)(const void*)(box_o + (size_t)b * NDET * 4);

        // D# group 0: count=1 (valid), lds_addr, global_addr, type=2 ("image")
        i32x4 g0;
        g0[0] = 1;
        g0[1] = (int)lds_addr;
        g0[2] = (int)(unsigned)(ga & 0xffffffffull);
        g0[3] = (int)(((unsigned)((ga >> 32) & 0x1ffffffull)) | (2u << 30));

        // D# group 1: data_size=4B, tensor_dim0=16384, tensor_dim1=1,
        // tile_dim0=16384, tile_dim1=1, tensor_dim0_stride=16384
        i32x8 g1;
        g1[0] = (int)(2u << 16);       // wg_mask=0, data_size=2 (4 bytes)
        g1[1] = (int)(16384u << 16);   // tensor_dim0[15:0] at bits[63:48]
        g1[2] = (int)(1u << 16);       // tensor_dim0[31:16]=0; tensor_dim1=1
        g1[3] = (int)(16384u << 16);   // tensor_dim1[31:16]=0; tile_dim0=16384
        g1[4] = 1;                     // tile_dim1=1, tile_dim2=0
        g1[5] = 16384;                 // tensor_dim0_stride[31:0]
        g1[6] = 0;
        g1[7] = 0;

        i32x4 gz4 = {0, 0, 0, 0};
#if defined(__clang_major__) && (__clang_major__ >= 23)
        i32x8 gz8 = {0, 0, 0, 0, 0, 0, 0, 0};
        __builtin_amdgcn_tensor_load_to_lds(g0, g1, gz4, gz4, gz8, 0);
#else
        __builtin_amdgcn_tensor_load_to_lds(g0, g1, gz4, gz4, 0);
#endif
        __builtin_amdgcn_s_wait_tensorcnt(0);
    }
    __syncthreads();

    // ---- Phase 1: decode boxes + scores into registers ----
    const float scl   = img_scale[b];
    const float inv_s = 1.0f / scl;
    const float clipx = img_size[b * 2 + 0] * inv_s;  // clamps cols 0,2 (x1,x2)
    const float clipy = img_size[b * 2 + 1] * inv_s;  // clamps cols 1,3 (y1,y2)

    float bx1[KPT], by1[KPT], bx2[KPT], by2[KPT], bar[KPT], bsc[KPT];
    int   bcl[KPT];
    unsigned validMask = (1u << KPT) - 1u;

    Sel v; v.s = -FLT_MAX; v.i = -1; v.x1 = v.y1 = v.x2 = v.y2 = 0.0f; v.c = 0;

    const int base = b * NDET;
#pragma unroll
    for (int k = 0; k < KPT; ++k) {
        const int i    = tid + k * NTHREADS;
        const int aidx = indices[base + i];
        const int cid  = classes[base + i];
        const float4 a = reinterpret_cast<const float4*>(anchors)[aidx];
        const float yca = (a.x + a.z) * 0.5f;
        const float xca = (a.y + a.w) * 0.5f;
        const float ha  = a.z - a.x;
        const float wa  = a.w - a.y;
        const float ty = rawBox[i * 4 + 0];
        const float tx = rawBox[i * 4 + 1];
        const float th = rawBox[i * 4 + 2];
        const float tw = rawBox[i * 4 + 3];
        const float w  = expf(tw) * wa;
        const float h  = expf(th) * ha;
        const float yc = ty * ha + yca;
        const float xc = tx * wa + xca;
        float x1 = xc - 0.5f * w, y1 = yc - 0.5f * h;
        float x2 = xc + 0.5f * w, y2 = yc + 0.5f * h;
        x1 = fminf(fmaxf(x1, 0.0f), clipx);
        y1 = fminf(fmaxf(y1, 0.0f), clipy);
        x2 = fminf(fmaxf(x2, 0.0f), clipx);
        y2 = fminf(fmaxf(y2, 0.0f), clipy);
        const float sc = 1.0f / (1.0f + expf(-cls_o[base + i]));
        bx1[k] = x1; by1[k] = y1; bx2[k] = x2; by2[k] = y2;
        bar[k] = fmaxf(x2 - x1, 0.0f) * fmaxf(y2 - y1, 0.0f);
        bsc[k] = sc; bcl[k] = cid;
        if (sc > v.s || (sc == v.s && (v.i < 0 || i < v.i))) {
            v.s = sc; v.i = i; v.x1 = x1; v.y1 = y1; v.x2 = x2; v.y2 = y2; v.c = cid;
        }
    }
    __syncthreads();  // done reading rawBox; its LDS now becomes reduce scratch

    float* sm  = reinterpret_cast<float*>(smem);
    int*   smi = reinterpret_cast<int*>(smem);

    // ---- Phase 2: greedy class-aware NMS, 100 steps ----
    blockReduceBroadcast(v, lane, wave, sm);  // selection for t=0

    float* dout = out + (size_t)b * MAXDET * 6;
    for (int t = 0; t < MAXDET; ++t) {
        const float selS = sm[112];
        const int   selI = smi[113];
        const float sx1 = sm[114], sy1 = sm[115], sx2 = sm[116], sy2 = sm[117];
        const int   selC = smi[118];
        const bool  anyV = (selI >= 0);

        if (tid == 0) {
            float* row = dout + t * 6;
            if (anyV) {
                row[0] = sx1 * scl;
                row[1] = sy1 * scl;
                row[2] = sx2 * scl;
                row[3] = sy2 * scl;
                row[4] = selS;
                row[5] = (float)(selC + 1);
            } else {
                row[0] = 0.0f; row[1] = 0.0f; row[2] = 0.0f;
                row[3] = 0.0f; row[4] = 0.0f; row[5] = 0.0f;
            }
        }
        if (t == MAXDET - 1) break;

        // Fused: suppress vs selected box (same class, IoU>thr) + next local argmax
        Sel nv; nv.s = -FLT_MAX; nv.i = -1; nv.x1 = nv.y1 = nv.x2 = nv.y2 = 0.0f; nv.c = 0;
        const float selArea = fmaxf(sx2 - sx1, 0.0f) * fmaxf(sy2 - sy1, 0.0f);
#pragma unroll
        for (int k = 0; k < KPT; ++k) {
            if (!(validMask & (1u << k))) continue;
            const int i = tid + k * NTHREADS;
            bool kill = false;
            if (anyV) {
                if (i == selI) {
                    kill = true;
                } else if (bcl[k] == selC) {
                    const float lx = fmaxf(bx1[k], sx1);
                    const float ly = fmaxf(by1[k], sy1);
                    const float rx = fminf(bx2[k], sx2);
                    const float ry = fminf(by2[k], sy2);
                    const float inter = fmaxf(rx - lx, 0.0f) * fmaxf(ry - ly, 0.0f);
                    const float uni   = bar[k] + selArea - inter;
                    kill = (inter / fmaxf(uni, 1e-8f)) > IOU_THR;
                }
            }
            if (kill) { validMask &= ~(1u << k); continue; }
            if (bsc[k] > nv.s || (bsc[k] == nv.s && (nv.i < 0 || i < nv.i))) {
                nv.s = bsc[k]; nv.i = i;
                nv.x1 = bx1[k]; nv.y1 = by1[k]; nv.x2 = bx2[k]; nv.y2 = by2[k];
                nv.c = bcl[k];
            }
        }
        blockReduceBroadcast(nv, lane, wave, sm);
    }
}

extern "C" void kernel_launch(void* const* d_in, const int* in_sizes, int n_in,
                              void* d_out, int out_size, void* d_ws, size_t ws_size,
                              hipStream_t stream) {
    const float* cls_o     = (const float*)d_in[0];  // (8, 4096, 1) f32
    const float* box_o     = (const float*)d_in[1];  // (8, 4096, 4) f32
    const int*   indices   = (const int*)d_in[2];    // (8, 4096) i32
    const int*   classes   = (const int*)d_in[3];    // (8, 4096) i32
    const float* anchors   = (const float*)d_in[4];  // (49104, 4) f32
    const float* img_scale = (const float*)d_in[5];  // (8,) f32
    const float* img_size  = (const float*)d_in[6];  // (8, 2) f32
    float* outp = (float*)d_out;                     // (8, 100, 6) f32

    hipLaunchKernelGGL(effdet_nms_kernel, dim3(NB), dim3(NTHREADS), 0, stream,
                       cls_o, box_o, indices, classes, anchors, img_scale,
                       img_size, outp);
}